// SVD_11682311045216
// MI455X (gfx1250) — compile-verified
//
#include <hip/hip_runtime.h>
#include <math.h>

#define NPTS 2048
#define TPB  256

typedef float v2f __attribute__((ext_vector_type(2)));
typedef float v8f __attribute__((ext_vector_type(8)));

// Per block (one batch): stream N points, accumulate 16 partial sums per lane,
// reduce 256x16 partials to 16 sums with V_WMMA_F32_16X16X4_F32 (A = ones),
// then thread 0 runs the 3x3 Kabsch SVD tail.
__global__ __launch_bounds__(TPB) void kabsch_wmma_kernel(
    const float* __restrict__ src, const float* __restrict__ tgt,
    const float* __restrict__ wgt, float* __restrict__ out, int B)
{
    __shared__ float P[TPB][17];   // 256 rows x 16 partials (padded: bank-conflict free)
    __shared__ float Q[8][16];     // per-wave partial sums
    __shared__ float F[16];        // final sums

    const int b    = blockIdx.x;
    const int tid  = threadIdx.x;
    const int lane = tid & 31;
    const int wv   = tid >> 5;

    const size_t baseC = (size_t)b * NPTS * 3;
    const size_t baseW = (size_t)b * NPTS;

    // 16 accumulators: [0]=Sum(w), [1..3]=Sum(w*s), [4..6]=Sum(w*t),
    // [7..15]=Sum(w * t_i * s_j) row-major (i,j)
    float aw = 0.f;
    float as0 = 0.f, as1 = 0.f, as2 = 0.f;
    float at0 = 0.f, at1 = 0.f, at2 = 0.f;
    float m00 = 0.f, m01 = 0.f, m02 = 0.f;
    float m10 = 0.f, m11 = 0.f, m12 = 0.f;
    float m20 = 0.f, m21 = 0.f, m22 = 0.f;

#pragma unroll
    for (int i = 0; i < NPTS / TPB; ++i) {
        const int p = i * TPB + tid;                 // coalesced: lanes adjacent
        const float* sp = src + baseC + 3 * p;       // b96 per lane, contiguous wave
        const float* tp = tgt + baseC + 3 * p;
        const float sx = sp[0], sy = sp[1], sz = sp[2];
        const float tx = tp[0], ty = tp[1], tz = tp[2];
        const float w  = wgt[baseW + p];
        const float wx = w * tx, wy = w * ty, wz = w * tz;
        aw += w;
        as0 = fmaf(w, sx, as0); as1 = fmaf(w, sy, as1); as2 = fmaf(w, sz, as2);
        at0 += wx; at1 += wy; at2 += wz;
        m00 = fmaf(wx, sx, m00); m01 = fmaf(wx, sy, m01); m02 = fmaf(wx, sz, m02);
        m10 = fmaf(wy, sx, m10); m11 = fmaf(wy, sy, m11); m12 = fmaf(wy, sz, m12);
        m20 = fmaf(wz, sx, m20); m21 = fmaf(wz, sy, m21); m22 = fmaf(wz, sz, m22);
    }

    P[tid][0]  = aw;
    P[tid][1]  = as0; P[tid][2]  = as1; P[tid][3]  = as2;
    P[tid][4]  = at0; P[tid][5]  = at1; P[tid][6]  = at2;
    P[tid][7]  = m00; P[tid][8]  = m01; P[tid][9]  = m02;
    P[tid][10] = m10; P[tid][11] = m11; P[tid][12] = m12;
    P[tid][13] = m20; P[tid][14] = m21; P[tid][15] = m22;
    __syncthreads();

#if __has_builtin(__builtin_amdgcn_wmma_f32_16x16x4_f32)
    {
        // A = ones(16x4): per ISA layout every lane holds {1,1}
        v2f a; a[0] = 1.f; a[1] = 1.f;
        // B(4x16) layout: lanes 0-15 -> (K=0,K=1) of column L; lanes 16-31 -> (K=2,K=3)
        const int kb  = (lane >> 4) << 1;
        const int col = lane & 15;

        // Stage 2: each wave reduces its own 32x16 stripe (8 chained WMMAs)
        v8f acc = {};
#pragma unroll
        for (int c = 0; c < 8; ++c) {
            const int r0 = wv * 32 + c * 4;
            v2f bb;
            bb[0] = P[r0 + kb][col];
            bb[1] = P[r0 + kb + 1][col];
            acc = __builtin_amdgcn_wmma_f32_16x16x4_f32(
                false, a, false, bb, (short)0, acc, false, false);
        }
        // D row M=0 lives in VGPR0 of lanes 0-15: D[0][L] = stripe column sum
        if (lane < 16) Q[wv][lane] = acc[0];
        __syncthreads();

        // Stage 3: wave 0 folds the 8 wave partials (2 WMMAs); wave-uniform branch
        if (wv == 0) {
            v8f f2 = {};
#pragma unroll
            for (int c = 0; c < 2; ++c) {
                const int r0 = c * 4;
                v2f bb;
                bb[0] = Q[r0 + kb][col];
                bb[1] = Q[r0 + kb + 1][col];
                f2 = __builtin_amdgcn_wmma_f32_16x16x4_f32(
                    false, a, false, bb, (short)0, f2, false, false);
            }
            if (lane < 16) F[lane] = f2[0];
        }
        __syncthreads();
    }
#else
    if (tid < 16) {
        float s = 0.f;
        for (int r = 0; r < TPB; ++r) s += P[r][tid];
        F[tid] = s;
    }
    __syncthreads();
#endif

    if (tid == 0) {
        const float Swt = F[0];
        const float w   = Swt + 1e-4f;
        const float Ssv[3] = {F[1], F[2], F[3]};
        const float Stv[3] = {F[4], F[5], F[6]};
        float sc[3], tc[3];
        for (int i = 0; i < 3; ++i) { sc[i] = Ssv[i] / w; tc[i] = Stv[i] / w; }

        // Centered weighted cross-covariance from uncentered sums
        float C[3][3];
        for (int i = 0; i < 3; ++i)
            for (int j = 0; j < 3; ++j)
                C[i][j] = (F[7 + 3 * i + j] - Stv[i] * sc[j] - tc[i] * Ssv[j]
                           + Swt * tc[i] * sc[j]) / w;

        // A = C^T C, symmetric PSD; Jacobi eigendecomposition A = V diag(lam) V^T
        float A[3][3];
        for (int j = 0; j < 3; ++j)
            for (int k = 0; k < 3; ++k)
                A[j][k] = C[0][j] * C[0][k] + C[1][j] * C[1][k] + C[2][j] * C[2][k];

        float V[3][3] = {{1.f, 0.f, 0.f}, {0.f, 1.f, 0.f}, {0.f, 0.f, 1.f}};
        for (int sweep = 0; sweep < 10; ++sweep) {
            for (int pair = 0; pair < 3; ++pair) {
                const int p = (pair == 2) ? 1 : 0;
                const int q = (pair == 0) ? 1 : 2;
                const float apq = A[p][q];
                if (fabsf(apq) > 1e-24f) {
                    const float app = A[p][p], aqq = A[q][q];
                    const float tau = (aqq - app) / (2.f * apq);
                    const float tsg = (tau >= 0.f) ? 1.f : -1.f;
                    const float t   = tsg / (fabsf(tau) + sqrtf(1.f + tau * tau));
                    const float cc  = 1.f / sqrtf(1.f + t * t);
                    const float ss  = t * cc;
                    const int   r   = 3 - p - q;
                    const float arp = A[r][p], arq = A[r][q];
                    A[p][p] = app - t * apq;
                    A[q][q] = aqq + t * apq;
                    A[p][q] = 0.f; A[q][p] = 0.f;
                    A[r][p] = cc * arp - ss * arq; A[p][r] = A[r][p];
                    A[r][q] = ss * arp + cc * arq; A[q][r] = A[r][q];
                    for (int k = 0; k < 3; ++k) {
                        const float vp = V[k][p], vq = V[k][q];
                        V[k][p] = cc * vp - ss * vq;
                        V[k][q] = ss * vp + cc * vq;
                    }
                }
            }
        }
        float lam[3] = {A[0][0], A[1][1], A[2][2]};
        // sort eigenpairs descending (det-fix must hit smallest sigma)
        for (int a1 = 0; a1 < 2; ++a1)
            for (int a2 = 0; a2 < 2 - a1; ++a2)
                if (lam[a2] < lam[a2 + 1]) {
                    const float tl = lam[a2]; lam[a2] = lam[a2 + 1]; lam[a2 + 1] = tl;
                    for (int k = 0; k < 3; ++k) {
                        const float tv = V[k][a2];
                        V[k][a2] = V[k][a2 + 1]; V[k][a2 + 1] = tv;
                    }
                }

        const float detC =
            C[0][0] * (C[1][1] * C[2][2] - C[1][2] * C[2][1]) -
            C[0][1] * (C[1][0] * C[2][2] - C[1][2] * C[2][0]) +
            C[0][2] * (C[1][0] * C[2][1] - C[1][1] * C[2][0]);
        const float s3 = (detC < 0.f) ? -1.f : 1.f;   // det(U)det(V) = sign(det W)
        const float d0 = 1.f / sqrtf(fmaxf(lam[0], 1e-30f));
        const float d1 = 1.f / sqrtf(fmaxf(lam[1], 1e-30f));
        const float d2 = s3 / sqrtf(fmaxf(lam[2], 1e-30f));

        // R = C * (V diag(d) V^T)
        float Bm[3][3];
        for (int j = 0; j < 3; ++j)
            for (int k = 0; k < 3; ++k)
                Bm[j][k] = V[j][0] * d0 * V[k][0] + V[j][1] * d1 * V[k][1]
                         + V[j][2] * d2 * V[k][2];
        float R[3][3];
        for (int i = 0; i < 3; ++i)
            for (int k = 0; k < 3; ++k)
                R[i][k] = C[i][0] * Bm[0][k] + C[i][1] * Bm[1][k] + C[i][2] * Bm[2][k];

        float* Ro = out + (size_t)b * 9;
        for (int i = 0; i < 3; ++i)
            for (int j = 0; j < 3; ++j)
                Ro[3 * i + j] = R[i][j];

        // t_src_tgt_intgt = tc - R * sc   (R orthogonal)
        float* To = out + (size_t)B * 9 + (size_t)b * 3;
        for (int i = 0; i < 3; ++i)
            To[i] = tc[i] - (R[i][0] * sc[0] + R[i][1] * sc[1] + R[i][2] * sc[2]);
    }
}

extern "C" void kernel_launch(void* const* d_in, const int* in_sizes, int n_in,
                              void* d_out, int out_size, void* d_ws, size_t ws_size,
                              hipStream_t stream) {
    const float* src = (const float*)d_in[0];
    const float* tgt = (const float*)d_in[1];
    const float* wgt = (const float*)d_in[2];
    float* out = (float*)d_out;
    const int B = in_sizes[0] / (3 * NPTS);
    kabsch_wmma_kernel<<<dim3(B), dim3(TPB), 0, stream>>>(src, tgt, wgt, out, B);
}